// TripletLoss_2525440770462
// MI455X (gfx1250) — compile-verified
//
#include <hip/hip_runtime.h>
#include <math.h>

typedef __attribute__((ext_vector_type(2))) float v2f;
typedef __attribute__((ext_vector_type(8))) float v8f;
typedef __attribute__((ext_vector_type(4))) int vi4;

#define N 8192
#define D 128
#define ROWS 32        // rows per block (2 waves x 16 rows)
#define BN 64          // column tile width (4 WMMA subtiles)
#define NSUB (BN / 16)
#define LDK 132        // padded K stride in LDS floats
#define CSPLIT 4       // column-dimension split across blocks (gridDim.y)
#define CCHUNK (N / CSPLIT)
#define MARGIN 0.3f
#define POS_SHIFT 16.0f  // fixed softmax shift for positive logits (cancels in t/s)

// ---- CDNA5 helpers (compile-safe fallbacks) --------------------------------

__device__ __forceinline__ float fast_sqrtf(float x) {
#if __has_builtin(__builtin_amdgcn_sqrtf)
    return __builtin_amdgcn_sqrtf(x);   // raw v_sqrt_f32
#else
    return sqrtf(x);
#endif
}

// 16B global -> LDS copy; async (no VGPR round-trip) when the gfx1250 builtin exists.
__device__ __forceinline__ void copy16_to_lds(float* lds, const float* g) {
#if __has_builtin(__builtin_amdgcn_global_load_async_to_lds_b128)
    __builtin_amdgcn_global_load_async_to_lds_b128(
        (__attribute__((address_space(1))) vi4*)(vi4*)(void*)(const_cast<float*>(g)),
        (__attribute__((address_space(3))) vi4*)(vi4*)(void*)(lds),
        0, 0);
#else
    *(float4*)lds = *(const float4*)g;
#endif
}

__device__ __forceinline__ void wait_async_copies() {
#if __has_builtin(__builtin_amdgcn_global_load_async_to_lds_b128)
#if __has_builtin(__builtin_amdgcn_s_wait_asynccnt)
    __builtin_amdgcn_s_wait_asynccnt(0);
#elif defined(__gfx1250__)
    asm volatile("s_wait_asynccnt 0x0" ::: "memory");
#endif
#endif
}

// ---- Kernels ---------------------------------------------------------------

__global__ void norms_kernel(const float* __restrict__ feat, float* __restrict__ x2) {
    int gid = blockIdx.x * blockDim.x + threadIdx.x;
    int row = gid >> 5;
    int lane = threadIdx.x & 31;
    if (row >= N) return;
    float4 v = *(const float4*)&feat[row * D + lane * 4];
    float s = v.x * v.x + v.y * v.y + v.z * v.z + v.w * v.w;
    #pragma unroll
    for (int off = 16; off > 0; off >>= 1)
        s += __shfl_xor(s, off, 32);
    if (lane == 0) x2[row] = s;
}

__global__ void __launch_bounds__(64)
triplet_main_kernel(const float* __restrict__ feat,
                    const int* __restrict__ labels,
                    const float* __restrict__ x2,
                    float4* __restrict__ partials) {   // [row * CSPLIT + chunk]
    __shared__ float As[ROWS * LDK];
    __shared__ float Bs[BN * LDK];
    __shared__ float colX2S[BN];
    __shared__ int colLabS[BN];

    const int tid = threadIdx.x;
    const int lane = tid & 31;
    const int wave = tid >> 5;          // 0..1
    const int lo = lane & 15;
    const int hi = lane >> 4;
    const int rowBase = blockIdx.x * ROWS;
    const int waveRow = rowBase + wave * 16;
    const int chunk = blockIdx.y;
    const int colStart = chunk * CCHUNK;

    // Stage A stripe: 32 rows x 128 K (waited via first in-loop wait/barrier)
    for (int idx = tid * 4; idx < ROWS * D; idx += 64 * 4) {
        int r = idx >> 7;
        int k = idx & (D - 1);
        copy16_to_lds(&As[r * LDK + k], &feat[(rowBase + r) * D + k]);
    }

    int labRow[8];
    float x2Row[8];
    #pragma unroll
    for (int g = 0; g < 8; ++g) {
        int r = waveRow + g + 8 * hi;
        labRow[g] = labels[r];
        x2Row[g] = x2[r];
    }

    // Fixed-shift softmax accumulators; per lane covers cols == lo (mod 16).
    float sp[8], tp[8], sn[8], tn[8];
    #pragma unroll
    for (int g = 0; g < 8; ++g) { sp[g] = 0.f; tp[g] = 0.f; sn[g] = 0.f; tn[g] = 0.f; }

    for (int colBase = colStart; colBase < colStart + CCHUNK; colBase += BN) {
        __syncthreads();   // previous tile's consumers done (covers As staging on iter 0)
        for (int idx = tid * 4; idx < BN * D; idx += 64 * 4) {
            int r = idx >> 7;
            int k = idx & (D - 1);
            copy16_to_lds(&Bs[r * LDK + k], &feat[(colBase + r) * D + k]);
        }
        colLabS[tid] = labels[colBase + tid];          // blockDim == BN == 64
        colX2S[tid] = x2[colBase + tid];
        wait_async_copies();
        __syncthreads();

        v8f c[NSUB];
        #pragma unroll
        for (int s = 0; s < NSUB; ++s) c[s] = (v8f){};

        const float* aRow = &As[(wave * 16 + lo) * LDK + 2 * hi];
        const float* bRow = &Bs[lo * LDK + 2 * hi];
        #pragma unroll 4
        for (int kk = 0; kk < D; kk += 4) {
            v2f a = *(const v2f*)(aRow + kk);
            #pragma unroll
            for (int s = 0; s < NSUB; ++s) {
                v2f b = *(const v2f*)(bRow + s * 16 * LDK + kk);
                c[s] = __builtin_amdgcn_wmma_f32_16x16x4_f32(false, a, false, b,
                                                             (short)0, c[s], false, false);
            }
        }

        // Element phase: branchless masked accumulation (one v_exp per element)
        #pragma unroll
        for (int s = 0; s < NSUB; ++s) {
            int col = colBase + s * 16 + lo;
            int labC = colLabS[s * 16 + lo];
            float x2C = colX2S[s * 16 + lo];
            #pragma unroll
            for (int g = 0; g < 8; ++g) {
                int row = waveRow + g + 8 * hi;
                float d2 = fmaf(-2.0f, c[s][g], x2Row[g] + x2C);
                float d = fast_sqrtf(fmaxf(d2, 1e-8f));
                bool same = (labC == labRow[g]);
                float logit = same ? (d - POS_SHIFT) : -d;
                float e = __expf(logit);
                float ep = (same && (row != col)) ? e : 0.f;   // positive weight
                float en = same ? 0.f : e;                      // negative weight
                sp[g] += ep;
                tp[g] = fmaf(ep, d, tp[g]);
                sn[g] += en;
                tn[g] = fmaf(en, d, tn[g]);
            }
        }
    }

    // Sum partials across the 16 lanes of each half (xor masks stay within half)
    #pragma unroll
    for (int g = 0; g < 8; ++g) {
        #pragma unroll
        for (int off = 1; off < 16; off <<= 1) {
            sp[g] += __shfl_xor(sp[g], off, 32);
            tp[g] += __shfl_xor(tp[g], off, 32);
            sn[g] += __shfl_xor(sn[g], off, 32);
            tn[g] += __shfl_xor(tn[g], off, 32);
        }
    }

    if (lo == 0) {  // lanes 0 (rows g) and 16 (rows g+8)
        #pragma unroll
        for (int g = 0; g < 8; ++g) {
            int row = waveRow + g + 8 * hi;
            partials[row * CSPLIT + chunk] = make_float4(sp[g], tp[g], sn[g], tn[g]);
        }
    }
}

__global__ void finalize_rows_kernel(const float4* __restrict__ partials,
                                     float* __restrict__ rowLoss,
                                     float* __restrict__ rowValid) {
    int row = blockIdx.x * blockDim.x + threadIdx.x;
    if (row >= N) return;
    float sp = 0.f, tp = 0.f, sn = 0.f, tn = 0.f;
    #pragma unroll
    for (int c = 0; c < CSPLIT; ++c) {
        float4 p = partials[row * CSPLIT + c];
        sp += p.x; tp += p.y; sn += p.z; tn += p.w;
    }
    bool valid = (sp > 0.f) && (sn > 0.f);
    float wp = (sp > 0.f) ? tp / sp : 0.f;
    float wn = (sn > 0.f) ? tn / sn : 0.f;
    float x = wp - wn + MARGIN;
    float l = (x > 20.f) ? x : log1pf(expf(x));   // softplus
    rowLoss[row] = valid ? l : 0.f;
    rowValid[row] = valid ? 1.f : 0.f;
}

__global__ void reduce_loss_kernel(const float* __restrict__ rowLoss,
                                   const float* __restrict__ rowValid,
                                   float* __restrict__ out) {
    __shared__ float sl[256];
    __shared__ float sv[256];
    float accL = 0.f, accV = 0.f;
    for (int i = threadIdx.x; i < N; i += 256) {
        accL += rowLoss[i];
        accV += rowValid[i];
    }
    sl[threadIdx.x] = accL;
    sv[threadIdx.x] = accV;
    __syncthreads();
    for (int s = 128; s > 0; s >>= 1) {
        if (threadIdx.x < s) {
            sl[threadIdx.x] += sl[threadIdx.x + s];
            sv[threadIdx.x] += sv[threadIdx.x + s];
        }
        __syncthreads();
    }
    if (threadIdx.x == 0)
        out[0] = sl[0] / fmaxf(sv[0], 1.0f);
}

extern "C" void kernel_launch(void* const* d_in, const int* in_sizes, int n_in,
                              void* d_out, int out_size, void* d_ws, size_t ws_size,
                              hipStream_t stream) {
    const float* feat = (const float*)d_in[0];
    const int* labels = (const int*)d_in[1];
    float* out = (float*)d_out;

    float* x2 = (float*)d_ws;              // N floats
    float* rowLoss = x2 + N;               // N floats
    float* rowValid = rowLoss + N;         // N floats
    float4* partials = (float4*)(rowValid + N);  // N * CSPLIT float4 (16B aligned)

    norms_kernel<<<(N * 32) / 256, 256, 0, stream>>>(feat, x2);
    dim3 grid(N / ROWS, CSPLIT);
    triplet_main_kernel<<<grid, 64, 0, stream>>>(feat, labels, x2, partials);
    finalize_rows_kernel<<<N / 256, 256, 0, stream>>>(partials, rowLoss, rowValid);
    reduce_loss_kernel<<<1, 256, 0, stream>>>(rowLoss, rowValid, out);
}